// mask_moe_10436770529969
// MI455X (gfx1250) — compile-verified
//
#include <hip/hip_runtime.h>
#include <hip/hip_bf16.h>
#include <stdint.h>

// ---------------------------------------------------------------------------
// Problem constants (from reference): B=32 H=8 L=512 E=3 N_VARS=8 -> N=64
// ---------------------------------------------------------------------------
#define L_DIM      512
#define E_EXP      3
#define BH         256                  // B*H
#define ROWS       (BH * L_DIM)         // 131072 rows of xf
#define WAVES_PB   8
#define THREADS_PB (WAVES_PB * 32)      // 256
#define NUM_BLOCKS (ROWS / (16 * WAVES_PB))  // 1024 (each wave = one 16-row tile)
#define WPAD_STR   516                  // LDS row stride for B operand (bank spread)
#define CT_STR     17                   // padded 16x16 C tile stride
#define IMP_N      (L_DIM * E_EXP)      // 1536 importance accumulators
#define WS_ENT_OFF IMP_N                // entropy accumulator slot in ws
#define OUT_ELEMS  ((size_t)32 * 8 * 512 * 512)  // 67108864; loss follows

#define TOP_P      0.5f
#define NOISE_EPS  0.01f
#define EPS_LOG    1e-10f

typedef __attribute__((ext_vector_type(2))) float v2f;
typedef __attribute__((ext_vector_type(4))) float v4f;
typedef __attribute__((ext_vector_type(8))) float v8f;

// ---------------------------------------------------------------------------
// Deterministic counter-based RNG (stand-in for jax threefry; no runtime check)
// ---------------------------------------------------------------------------
__device__ __forceinline__ unsigned pcg(unsigned x) {
  x ^= x >> 17; x *= 0xed5ad4bbu;
  x ^= x >> 11; x *= 0xac4c1b51u;
  x ^= x >> 15; x *= 0x31848babu;
  x ^= x >> 14;
  return x;
}

__device__ __forceinline__ float gauss_noise(unsigned idx) {
  // Box-Muller from two hashed uniforms
  float u1 = (pcg(idx * 2u + 0x9e3779b9u) + 0.5f) * 2.3283064365386963e-10f;
  float u2 = (pcg(idx * 2u + 0x7f4a7c15u) + 0.5f) * 2.3283064365386963e-10f;
  u1 = fmaxf(u1, 1e-9f);
  return __fsqrt_rn(-2.0f * __logf(u1)) * __cosf(6.2831853071795864f * u2);
}

__device__ __forceinline__ float softplus_f(float v) {
  return v > 20.0f ? v : log1pf(__expf(v));
}

// ---------------------------------------------------------------------------
// Fused kernel: WMMA logits -> noisy top-p gating -> masked-broadcast output.
// One wave handles a 16-row tile of xf = x.reshape(BH, L, L).
// Both x (256MB) and out (256MB) are single-touch streams larger than the
// 192MB L2 -> access them with non-temporal cache policy.
// ---------------------------------------------------------------------------
__global__ __launch_bounds__(THREADS_PB) void moe_main(
    const float* __restrict__ x, const float* __restrict__ gate_w,
    const float* __restrict__ noise_w, float* __restrict__ out,
    float* __restrict__ ws) {
  __shared__ float Wpad[16 * WPAD_STR];              // B operand: 16 padded weight cols
  __shared__ float ctile[WAVES_PB][16 * CT_STR];     // per-wave 16x16 WMMA result
  __shared__ float gtile[WAVES_PB][16][4];           // per-wave gates for 16 rows
  __shared__ float entAcc;                           // block entropy accumulator

  const int tid  = threadIdx.x;
  const int lane = tid & 31;
  const int wave = tid >> 5;

  // --- Stage 0: build padded weight matrix in LDS (cols 0-2 gate, 3-5 noise) ---
  if (tid == 0) entAcc = 0.0f;
  for (int idx = tid; idx < 16 * L_DIM; idx += THREADS_PB) {
    int n = idx >> 9;          // 0..15
    int k = idx & (L_DIM - 1); // 0..511
    float v = 0.0f;
    if (n < E_EXP)            v = gate_w[n * L_DIM + k];
    else if (n < 2 * E_EXP)   v = noise_w[(n - E_EXP) * L_DIM + k];
    Wpad[n * WPAD_STR + k] = v;
  }
  __syncthreads();

  // --- Stage 1: logits via V_WMMA_F32_16X16X4_F32, accumulate over K=512 ---
  const int tile_row0 = (blockIdx.x * WAVES_PB + wave) * 16;   // first xf row
  // A fragment addressing: lane holds row m=lane&15, K pair (lane>>4)*2
  const size_t a_base =
      (size_t)(tile_row0 + (lane & 15)) * L_DIM + (size_t)((lane >> 4) * 2);
  // B fragment addressing: lane holds col n=lane&15, K pair (lane>>4)*2
  const int b_base = (lane & 15) * WPAD_STR + (lane >> 4) * 2;

  v8f acc = {0.f, 0.f, 0.f, 0.f, 0.f, 0.f, 0.f, 0.f};
#pragma unroll 8
  for (int k0 = 0; k0 < L_DIM; k0 += 4) {
    v2f a = __builtin_nontemporal_load((const v2f*)(x + a_base + k0));
    v2f b = *(const v2f*)(Wpad + b_base + k0);
    acc = __builtin_amdgcn_wmma_f32_16x16x4_f32(
        /*neg_a=*/false, a, /*neg_b=*/false, b,
        /*c_mod=*/(short)0, acc, /*reuse_a=*/false, /*reuse_b=*/false);
  }

  // Spill C tile to LDS: VGPR r holds M = 8*(lane>=16)+r, N = lane&15
  {
    float* ct = ctile[wave];
    const int mBase = (lane >> 4) * 8;
    const int n = lane & 15;
#pragma unroll
    for (int r = 0; r < 8; ++r) ct[(mBase + r) * CT_STR + n] = acc[r];
  }
  __syncthreads();

  // --- Stage 2: per-row noisy top-p gating (lanes 0..15 each own one row) ---
  if (lane < 16) {
    const float* ct = ctile[wave] + lane * CT_STR;
    const int row = tile_row0 + lane;     // = bh*512 + l
    const int l = row & (L_DIM - 1);

    float noisy[E_EXP];
#pragma unroll
    for (int e = 0; e < E_EXP; ++e) {
      float clean = ct[e];
      float stddev = softplus_f(ct[E_EXP + e]) + NOISE_EPS;
      noisy[e] = fmaf(gauss_noise((unsigned)(row * E_EXP + e)), stddev, clean);
    }
    // softmax over 3 experts
    float mx = fmaxf(noisy[0], fmaxf(noisy[1], noisy[2]));
    float p0 = __expf(noisy[0] - mx), p1 = __expf(noisy[1] - mx),
          p2 = __expf(noisy[2] - mx);
    float inv = 1.0f / (p0 + p1 + p2);
    p0 *= inv; p1 *= inv; p2 *= inv;

    // entropy contribution (negated): -sum p log(p+eps)
    float ent = -(p0 * __logf(p0 + EPS_LOG) + p1 * __logf(p1 + EPS_LOG) +
                  p2 * __logf(p2 + EPS_LOG));
    atomicAdd(&entAcc, ent);   // ds_add_f32

    // sort 3 descending with index tracking
    float s[3] = {p0, p1, p2};
    int id[3] = {0, 1, 2};
    if (s[0] < s[1]) { float t = s[0]; s[0] = s[1]; s[1] = t; int u = id[0]; id[0] = id[1]; id[1] = u; }
    if (s[0] < s[2]) { float t = s[0]; s[0] = s[2]; s[2] = t; int u = id[0]; id[0] = id[2]; id[2] = u; }
    if (s[1] < s[2]) { float t = s[1]; s[1] = s[2]; s[2] = t; int u = id[1]; id[1] = id[2]; id[2] = u; }

    float c0 = s[0], c1 = s[0] + s[1];
    bool m0 = c0 > TOP_P, m1 = c1 > TOP_P, m2 = (c1 + s[2]) > TOP_P;
    int thr = m0 ? 0 : (m1 ? 1 : 2);   // first crossing expert stays kept
    if (thr == 0) m0 = false; else if (thr == 1) m1 = false; else m2 = false;

    // gates scattered back to original expert ids
    float g[3];
    g[id[0]] = m0 ? 0.0f : 1.0f;
    g[id[1]] = m1 ? 0.0f : 1.0f;
    g[id[2]] = m2 ? 0.0f : 1.0f;
    gtile[wave][lane][0] = g[0];
    gtile[wave][lane][1] = g[1];
    gtile[wave][lane][2] = g[2];

    // importance over *sorted* positions, summed over batch
    atomicAdd(&ws[l * E_EXP + 0], m0 ? 0.0f : s[0]);
    atomicAdd(&ws[l * E_EXP + 1], m1 ? 0.0f : s[1]);
    atomicAdd(&ws[l * E_EXP + 2], m2 ? 0.0f : s[2]);
  }
  __syncthreads();
  if (tid == 0) atomicAdd(&ws[WS_ENT_OFF], entAcc);

  // --- Stage 3: write 16 output rows; masks collapse to a 3-way select ------
  // out[row, d] = (d/64==l/64) ? g1 : ((d%64==l%64) ? g0 : g2)   (+1 if d==l)
#pragma unroll
  for (int m = 0; m < 16; ++m) {
    const int row = tile_row0 + m;
    const int l = row & (L_DIM - 1);
    const int lb = l >> 6, lo = l & 63;
    const float g0 = gtile[wave][m][0];
    const float g1 = gtile[wave][m][1];
    const float g2 = gtile[wave][m][2];
    float* rowp = out + (size_t)row * L_DIM;
#pragma unroll
    for (int i = 0; i < 4; ++i) {
      const int d0 = (i * 32 + lane) * 4;
      const int db = d0 >> 6;           // whole float4 lies in one 64-block
      v4f v;
#pragma unroll
      for (int j = 0; j < 4; ++j) {
        const int d = d0 + j;
        float val;
        if (db == lb)            val = g1;
        else if ((d & 63) == lo) val = g0;
        else                     val = g2;
        if (d == l) val += 1.0f;
        v[j] = val;
      }
      __builtin_nontemporal_store(v, (v4f*)(rowp + d0));  // b128 NT stream store
    }
  }
}

// ---------------------------------------------------------------------------
// Loss finalize: var(importance, ddof=1)/(mean^2+eps) + 0.1 * ent/(BH*E)
// ---------------------------------------------------------------------------
__global__ __launch_bounds__(256) void moe_loss(const float* __restrict__ ws,
                                                float* __restrict__ loss_out) {
  __shared__ float ssum[256];
  __shared__ float ssq[256];
  const int tid = threadIdx.x;
  float sum = 0.0f, sq = 0.0f;
  for (int i = tid; i < IMP_N; i += 256) {
    float v = ws[i];
    sum += v;
    sq = fmaf(v, v, sq);
  }
  ssum[tid] = sum;
  ssq[tid] = sq;
  __syncthreads();
  for (int s = 128; s > 0; s >>= 1) {
    if (tid < s) {
      ssum[tid] += ssum[tid + s];
      ssq[tid] += ssq[tid + s];
    }
    __syncthreads();
  }
  if (tid == 0) {
    const float n = (float)IMP_N;
    float total = ssum[0];
    float mean = total / n;
    float var = (ssq[0] - total * mean) / (n - 1.0f);   // ddof=1
    float loss_imp = var / (mean * mean + EPS_LOG);
    float loss_dyn = ws[WS_ENT_OFF] / (float)(BH * E_EXP);
    loss_out[0] = loss_imp + 0.1f * loss_dyn;
  }
}

// ---------------------------------------------------------------------------
extern "C" void kernel_launch(void* const* d_in, const int* in_sizes, int n_in,
                              void* d_out, int out_size, void* d_ws,
                              size_t ws_size, hipStream_t stream) {
  const float* x  = (const float*)d_in[0];
  const float* gw = (const float*)d_in[1];
  const float* nw = (const float*)d_in[2];
  float* out = (float*)d_out;
  float* ws  = (float*)d_ws;

  // zero importance[1536] + entropy accumulator (capture-legal async memset)
  hipMemsetAsync(d_ws, 0, (IMP_N + 1) * sizeof(float), stream);

  moe_main<<<NUM_BLOCKS, THREADS_PB, 0, stream>>>(x, gw, nw, out, ws);
  moe_loss<<<1, 256, 0, stream>>>(ws, out + OUT_ELEMS);
}